// NarrowParabolicEq_10599979287223
// MI455X (gfx1250) — compile-verified
//
#include <hip/hip_runtime.h>

// ---------------- static problem configuration (mirrors the reference) ----
#define NPTS 500      // grid points
#define VPT  4        // points per thread
#define NTHR 128      // 4 wave32s, one WGP
#define NACT 125      // active threads (125*4 = 500)
#define N_INNER 10    // RK steps per save interval (TS_SPACING/DT)
#define PAD  2        // left halo pad (keeps stage writes 16B aligned)
#define LDSN (NPTS + PAD + 2)   // 504 float2 slots per buffer

constexpr double PI_D   = 3.14159265358979323846;
constexpr float  C0f    = 1500.0f;
constexpr float  K0f    = (float)(2.0 * PI_D * 50.0 / 1500.0);
constexpr float  INV2K0 = (float)(1.0 / (2.0 * (2.0 * PI_D * 50.0 / 1500.0)));
constexpr float  EPSM   = 0.00737f;
constexpr float  DXf    = (float)(2000.0 / 499.0);
constexpr float  INVDX2 = (float)(1.0 / ((2000.0/499.0)*(2000.0/499.0)));

// Dopri5 coefficients pre-multiplied by DT = 0.1
constexpr float dA21 = (float)(0.1 * (1.0/5.0));
constexpr float dA31 = (float)(0.1 * (3.0/40.0));
constexpr float dA32 = (float)(0.1 * (9.0/40.0));
constexpr float dA41 = (float)(0.1 * (44.0/45.0));
constexpr float dA42 = (float)(0.1 * (-56.0/15.0));
constexpr float dA43 = (float)(0.1 * (32.0/9.0));
constexpr float dA51 = (float)(0.1 * (19372.0/6561.0));
constexpr float dA52 = (float)(0.1 * (-25360.0/2187.0));
constexpr float dA53 = (float)(0.1 * (64448.0/6561.0));
constexpr float dA54 = (float)(0.1 * (-212.0/729.0));
constexpr float dA61 = (float)(0.1 * (9017.0/3168.0));
constexpr float dA62 = (float)(0.1 * (-355.0/33.0));
constexpr float dA63 = (float)(0.1 * (46732.0/5247.0));
constexpr float dA64 = (float)(0.1 * (49.0/176.0));
constexpr float dA65 = (float)(0.1 * (-5103.0/18656.0));
constexpr float dB1  = (float)(0.1 * (35.0/384.0));
constexpr float dB3  = (float)(0.1 * (500.0/1113.0));
constexpr float dB4  = (float)(0.1 * (125.0/192.0));
constexpr float dB5  = (float)(0.1 * (-2187.0/6784.0));
constexpr float dB6  = (float)(0.1 * (11.0/84.0));

// CDNA5 split workgroup barrier: signal early (after LDS writes retire via the
// split DS counter), run shadow work, wait late. asm volatile is a scheduling
// boundary, so code placed between SIGNAL and WAIT executes in the barrier
// arrival shadow.
#define BAR_SIGNAL() asm volatile("s_wait_dscnt 0x0\ns_barrier_signal -1" ::: "memory")
#define BAR_WAIT()   asm volatile("s_barrier_wait -1" ::: "memory")

extern "C" __global__ void __launch_bounds__(NTHR, 1)
pe_dopri5(const float* __restrict__ params, float* __restrict__ out, int n_save)
{
    // double-buffered stage vector; 16B aligned so stage writes are ds_store_b128
    __shared__ __align__(16) float2 stg[2][LDSN];

    const int  t    = threadIdx.x;
    const int  base = t * VPT;
    const bool act  = (t < NACT);
    const bool not_first = (t != 0);        // j=0 is interior unless t==0
    const bool not_last  = (t != NACT - 1); // j=3 is interior unless t==124

    const float ref_c = params[0];   // 1500.0 (Munk reference speed)
    const float ref_d = params[1];   // 1300.0 (Munk reference depth)

    // Gaussian aperture constants (elev = 0 -> initial field purely real)
    const float ww  = sqrtf(2.0f * logf(2.0f)) /
                      (K0f * sinf((float)(PI_D / 180.0 * 3.0 * 0.5)));
    const float amp = 1.0f / (sqrtf((float)PI_D) * ww);
    const float iww = 1.0f / ww;

    float yr[VPT], yi[VPT], kh[VPT];

    #pragma unroll
    for (int j = 0; j < VPT; ++j) {
        const int   p  = base + j;
        const float z  = (float)p * DXf;
        // Munk profile -> heterogeneity coefficient K0*((C0/c)^2 - 1)
        const float zz = 2.0f * (z - ref_d) / ref_d;
        const float c  = ref_c * (1.0f + EPSM * (zz - 1.0f + expf(-zz)));
        const float rr = C0f / c;
        const bool  in = (p > 0) && (p < NPTS - 1);
        kh[j]   = in ? (K0f * (rr * rr - 1.0f)) : 0.0f;
        // y0
        const float a = (z - 100.0f) * iww;
        yr[j] = amp * expf(-a * a);
        yi[j] = 0.0f;
    }

    // zero the halo pads once (boundary k is select-zeroed anyway; this keeps
    // transient lap values finite)
    if (t == 0) {
        const float2 z2 = make_float2(0.0f, 0.0f);
        stg[0][0] = z2; stg[0][1] = z2; stg[0][LDSN-2] = z2; stg[0][LDSN-1] = z2;
        stg[1][0] = z2; stg[1][1] = z2; stg[1][LDSN-2] = z2; stg[1][LDSN-1] = z2;
    }

    // save row 0 = y0  (fire-and-forget global_store_b128s)
    if (act) {
        float4* o = (float4*)(out + (size_t)base * 2);
        o[0] = make_float4(yr[0], yi[0], yr[1], yi[1]);
        o[1] = make_float4(yr[2], yi[2], yr[3], yi[3]);
    }

    int buf = 0;

    // rhs(y) = (i/(2K0)) * lap(y) + i*K0*het*y ; zero at both domain boundaries.
    // Interior points (j=1,2) and the register-resident part of the edge
    // stencils are computed in the barrier shadow; only a two-fma chain per edge
    // component remains after the halo load.
    auto RHS = [&](const float* sre, const float* sim, float* kre, float* kim) {
        if (act) {
            float4* w = (float4*)&stg[buf][base + PAD];        // 16B aligned
            w[0] = make_float4(sre[0], sim[0], sre[1], sim[1]);
            w[1] = make_float4(sre[2], sim[2], sre[3], sim[3]);
        }
        BAR_SIGNAL();
        // ---------------- barrier-arrival shadow (registers only) ---------
        float hr[VPT], hi[VPT];
        #pragma unroll
        for (int j = 0; j < VPT; ++j) {
            hr[j] = -kh[j] * sim[j];
            hi[j] =  kh[j] * sre[j];
        }
        // fully interior points: complete RHS now
        #pragma unroll
        for (int j = 1; j <= 2; ++j) {
            const float lre = (sre[j-1] - 2.0f * sre[j] + sre[j+1]) * INVDX2;
            const float lim = (sim[j-1] - 2.0f * sim[j] + sim[j+1]) * INVDX2;
            kre[j] = fmaf(-INV2K0, lim, hr[j]);
            kim[j] = fmaf( INV2K0, lre, hi[j]);
        }
        // edge partials: lap_edge = halo*INVDX2 + partial
        const float p0r = (sre[1] - 2.0f * sre[0]) * INVDX2;
        const float p0i = (sim[1] - 2.0f * sim[0]) * INVDX2;
        const float p3r = (sre[2] - 2.0f * sre[3]) * INVDX2;
        const float p3i = (sim[2] - 2.0f * sim[3]) * INVDX2;
        // ------------------------------------------------------------------
        BAR_WAIT();
        if (act) {
            const float2 lf = stg[buf][base + PAD - 1];        // point base-1
            const float2 rg = stg[buf][base + PAD + VPT];      // point base+4
            const float l0re = fmaf(lf.x, INVDX2, p0r);
            const float l0im = fmaf(lf.y, INVDX2, p0i);
            const float l3re = fmaf(rg.x, INVDX2, p3r);
            const float l3im = fmaf(rg.y, INVDX2, p3i);
            const float c0r = fmaf(-INV2K0, l0im, hr[0]);
            const float c0i = fmaf( INV2K0, l0re, hi[0]);
            const float c3r = fmaf(-INV2K0, l3im, hr[3]);
            const float c3i = fmaf( INV2K0, l3re, hi[3]);
            kre[0] = not_first ? c0r : 0.0f;   // v_cndmask: NaN-safe boundary zero
            kim[0] = not_first ? c0i : 0.0f;
            kre[3] = not_last  ? c3r : 0.0f;
            kim[3] = not_last  ? c3i : 0.0f;
        } else {
            #pragma unroll
            for (int j = 0; j < VPT; ++j) { kre[j] = 0.0f; kim[j] = 0.0f; }
        }
        buf ^= 1;
    };

    float k1r[VPT], k1i[VPT], k2r[VPT], k2i[VPT], k3r[VPT], k3i[VPT];
    float k4r[VPT], k4i[VPT], k5r[VPT], k5i[VPT], k6r[VPT], k6i[VPT];
    float sr[VPT], si[VPT];

    for (int s = 1; s < n_save; ++s) {
        for (int step = 0; step < N_INNER; ++step) {
            RHS(yr, yi, k1r, k1i);
            #pragma unroll
            for (int j = 0; j < VPT; ++j) {
                sr[j] = fmaf(dA21, k1r[j], yr[j]);
                si[j] = fmaf(dA21, k1i[j], yi[j]);
            }
            RHS(sr, si, k2r, k2i);
            #pragma unroll
            for (int j = 0; j < VPT; ++j) {
                sr[j] = fmaf(dA32, k2r[j], fmaf(dA31, k1r[j], yr[j]));
                si[j] = fmaf(dA32, k2i[j], fmaf(dA31, k1i[j], yi[j]));
            }
            RHS(sr, si, k3r, k3i);
            #pragma unroll
            for (int j = 0; j < VPT; ++j) {
                sr[j] = fmaf(dA43, k3r[j], fmaf(dA42, k2r[j], fmaf(dA41, k1r[j], yr[j])));
                si[j] = fmaf(dA43, k3i[j], fmaf(dA42, k2i[j], fmaf(dA41, k1i[j], yi[j])));
            }
            RHS(sr, si, k4r, k4i);
            #pragma unroll
            for (int j = 0; j < VPT; ++j) {
                sr[j] = fmaf(dA54, k4r[j], fmaf(dA53, k3r[j], fmaf(dA52, k2r[j], fmaf(dA51, k1r[j], yr[j]))));
                si[j] = fmaf(dA54, k4i[j], fmaf(dA53, k3i[j], fmaf(dA52, k2i[j], fmaf(dA51, k1i[j], yi[j]))));
            }
            RHS(sr, si, k5r, k5i);
            #pragma unroll
            for (int j = 0; j < VPT; ++j) {
                sr[j] = fmaf(dA65, k5r[j], fmaf(dA64, k4r[j], fmaf(dA63, k3r[j], fmaf(dA62, k2r[j], fmaf(dA61, k1r[j], yr[j])))));
                si[j] = fmaf(dA65, k5i[j], fmaf(dA64, k4i[j], fmaf(dA63, k3i[j], fmaf(dA62, k2i[j], fmaf(dA61, k1i[j], yi[j])))));
            }
            RHS(sr, si, k6r, k6i);
            #pragma unroll
            for (int j = 0; j < VPT; ++j) {
                yr[j] = fmaf(dB6, k6r[j], fmaf(dB5, k5r[j], fmaf(dB4, k4r[j], fmaf(dB3, k3r[j], fmaf(dB1, k1r[j], yr[j])))));
                yi[j] = fmaf(dB6, k6i[j], fmaf(dB5, k5i[j], fmaf(dB4, k4i[j], fmaf(dB3, k3i[j], fmaf(dB1, k1i[j], yi[j])))));
            }
        }
        if (act) {
            float4* o = (float4*)(out + ((size_t)s * NPTS + base) * 2);
            o[0] = make_float4(yr[0], yi[0], yr[1], yi[1]);
            o[1] = make_float4(yr[2], yi[2], yr[3], yi[3]);
        }
    }
}

extern "C" void kernel_launch(void* const* d_in, const int* in_sizes, int n_in,
                              void* d_out, int out_size, void* d_ws, size_t ws_size,
                              hipStream_t stream) {
    // setup_inputs order: d_in[0] = ts (float32, count = n_save), d_in[1] = params [ref_c, ref_depth]
    const float* params = (const float*)d_in[1];
    float*       out    = (float*)d_out;          // complex64 rows as interleaved float pairs
    const int    n_save = in_sizes[0];            // 201
    pe_dopri5<<<dim3(1), dim3(NTHR), 0, stream>>>(params, out, n_save);
}